// EdgeLevelGNN_24747601560282
// MI455X (gfx1250) — compile-verified
//
#include <hip/hip_runtime.h>
#include <hip/hip_bf16.h>

typedef float v2f __attribute__((ext_vector_type(2)));
typedef float v8f __attribute__((ext_vector_type(8)));

#define WMMA_F32X4(a, b, c) \
    __builtin_amdgcn_wmma_f32_16x16x4_f32(false, (a), false, (b), (short)0, (c), false, false)

constexpr int N_  = 50000;
constexpr int E_  = 800000;
constexpr int D_  = 128;
constexpr int H_  = 128;
constexpr int EA_ = 8;
constexpr int OUT_ = 2;
constexpr float EPS_ = 1e-5f;

// ---------------------------------------------------------------------------
// Degree via float atomics (deg counts in-edges; +1 self loop added in dinv)
// ---------------------------------------------------------------------------
__global__ void deg_kernel(const int* __restrict__ dst, float* __restrict__ deg) {
    int e = blockIdx.x * blockDim.x + threadIdx.x;
    if (e < E_) atomicAdd(&deg[dst[e]], 1.0f);
}

__global__ void dinv_kernel(const float* __restrict__ deg, float* __restrict__ dinv) {
    int n = blockIdx.x * blockDim.x + threadIdx.x;
    if (n < N_) dinv[n] = rsqrtf(deg[n] + 1.0f);
}

// ---------------------------------------------------------------------------
// Repack a row-major weight W[K][Nc] into per-lane WMMA B-fragment order:
//   Wp[(q*(Nc/16)+t)*32 + lane] = { W[q*4+kh2][t*16+(lane&15)],
//                                   W[q*4+kh2+1][ ... ] },  kh2 = (lane>>4)*2
// so a B fragment is one coalesced global_load_b64 per lane.
// ---------------------------------------------------------------------------
__global__ void pack_b_kernel(const float* __restrict__ W, v2f* __restrict__ Wp,
                              int K, int Nc) {
    const int tid = blockIdx.x * blockDim.x + threadIdx.x;
    const int tiles = Nc >> 4;
    const int total = (K >> 2) * tiles * 32;
    if (tid >= total) return;
    const int lane = tid & 31;
    const int t = (tid >> 5) % tiles;
    const int q = tid / (tiles * 32);
    const int n = t * 16 + (lane & 15);
    const int k = q * 4 + ((lane >> 4) << 1);
    v2f b;
    b.x = W[(size_t)k * Nc + n];
    b.y = W[(size_t)(k + 1) * Nc + n];
    Wp[tid] = b;
}

// ---------------------------------------------------------------------------
// H = X @ W   (N x 128) @ (128 x 128), fp32 WMMA 16x16x4, packed B.
// One wave -> 16 output rows x 128 cols (8 C fragments).
// ---------------------------------------------------------------------------
__global__ __launch_bounds__(256) void gemm_node_kernel(
    const float* __restrict__ X, const v2f* __restrict__ Wp,
    float* __restrict__ Hout) {
    const int wave = threadIdx.x >> 5;
    const int lane = threadIdx.x & 31;
    const int tile = blockIdx.x * 8 + wave;     // wave-uniform
    const int row0 = tile * 16;
    if (row0 >= N_) return;                     // uniform exit, EXEC stays all-1

    const int m = lane & 15;
    const int kh2 = (lane >> 4) << 1;           // 0 or 2
    const float* arow = X + (size_t)(row0 + m) * D_ + kh2;

    v8f c[8] = {};
    for (int q = 0; q < 32; ++q) {              // K-steps of 4
        v2f a = *(const v2f*)(arow + q * 4);
        const v2f* bp = Wp + (size_t)q * 8 * 32 + lane;
#pragma unroll
        for (int t = 0; t < 8; ++t)
            c[t] = WMMA_F32X4(a, bp[t * 32], c[t]);
    }
#pragma unroll
    for (int t = 0; t < 8; ++t) {
        const int n = t * 16 + m;
#pragma unroll
        for (int j = 0; j < 8; ++j) {
            const int mm = row0 + j + 4 * kh2;  // j + 8*(lane>=16)
            Hout[(size_t)mm * H_ + n] = c[t][j];
        }
    }
}

// ---------------------------------------------------------------------------
// agg[dst] += h[src] * (dinv[src]*dinv[dst]) : one thread = 4 features
// ---------------------------------------------------------------------------
__global__ void scatter_kernel(const float* __restrict__ Hin,
                               const int* __restrict__ src,
                               const int* __restrict__ dst,
                               const float* __restrict__ dinv,
                               float* __restrict__ agg) {
    const int tid = blockIdx.x * blockDim.x + threadIdx.x;
    const int e = tid >> 5;
    if (e >= E_) return;
    const int f = (tid & 31) << 2;
    const int s = src[e], d = dst[e];
    const float nrm = dinv[s] * dinv[d];
    const float4 hv = *(const float4*)(Hin + (size_t)s * H_ + f);
    float* ap = agg + (size_t)d * H_ + f;
    atomicAdd(ap + 0, hv.x * nrm);
    atomicAdd(ap + 1, hv.y * nrm);
    atomicAdd(ap + 2, hv.z * nrm);
    atomicAdd(ap + 3, hv.w * nrm);
}

// ---------------------------------------------------------------------------
// x = relu(BN(agg + h*dinv^2 + bc))
// ---------------------------------------------------------------------------
__global__ void combine_kernel(const float* __restrict__ agg,
                               const float* __restrict__ Hlin,
                               const float* __restrict__ dinv,
                               const float* __restrict__ bc,
                               const float* __restrict__ gamma,
                               const float* __restrict__ beta,
                               const float* __restrict__ rmean,
                               const float* __restrict__ rvar,
                               float* __restrict__ xout) {
    const int tid = blockIdx.x * blockDim.x + threadIdx.x;
    if (tid >= N_ * H_) return;
    const int node = tid >> 7;
    const int f = tid & (H_ - 1);
    const float di = dinv[node];
    float v = agg[tid] + Hlin[tid] * (di * di) + bc[f];
    v = (v - rmean[f]) * rsqrtf(rvar[f] + EPS_) * gamma[f] + beta[f];
    xout[tid] = fmaxf(v, 0.0f);
}

// ---------------------------------------------------------------------------
// Fused edge head: [x[src] | x[dst] | ea] -> 128 -> 64 -> 2, fp32 WMMA.
// One wave per 32-edge tile (two 16-row halves share every B fragment).
// Per-wave LDS scratch: h1 = 32x132 floats, h2 = 32x68 floats (bank padding).
// Block = 4 waves = 128 edges; LDS = 100 KB / workgroup.
// ---------------------------------------------------------------------------
__global__ __launch_bounds__(128) void edge_head_kernel(
    const float* __restrict__ X,
    const int* __restrict__ src, const int* __restrict__ dst,
    const float* __restrict__ EAttr,
    const v2f* __restrict__ Wp1, const float* __restrict__ bm1,
    const v2f* __restrict__ Wp2, const float* __restrict__ bm2,
    const float* __restrict__ Wm3, const float* __restrict__ bm3,
    float* __restrict__ Out) {
    extern __shared__ float smem[];                 // 4 waves * 6400 floats
    const int wave = threadIdx.x >> 5;
    const int lane = threadIdx.x & 31;
    float* h1 = smem + wave * 6400;                 // 32 x 132
    float* h2 = h1 + 32 * 132;                      // 32 x 68

    const int e0 = (blockIdx.x * 4 + wave) * 32;    // E_ = 6250*4*32 exactly
    const int m = lane & 15;
    const int kh2 = (lane >> 4) << 1;               // 0 or 2
    const int elo = e0 + m, ehi = e0 + 16 + m;
    const float* p0s = X + (size_t)src[elo] * H_;
    const float* p0d = X + (size_t)dst[elo] * H_;
    const float* p0e = EAttr + (size_t)elo * EA_;
    const float* p1s = X + (size_t)src[ehi] * H_;
    const float* p1d = X + (size_t)dst[ehi] * H_;
    const float* p1e = EAttr + (size_t)ehi * EA_;

    // ---- layer 1: C1[32x128] = A[32x264] @ Wm1[264x128] ----
    v8f clo[8] = {}, chi[8] = {};
    for (int q = 0; q < 66; ++q) {
        const int k = q * 4 + kh2;                  // region select wave-uniform
        v2f a0, a1;
        if (k < H_) {
            a0 = *(const v2f*)(p0s + k);
            a1 = *(const v2f*)(p1s + k);
        } else if (k < 2 * H_) {
            a0 = *(const v2f*)(p0d + (k - H_));
            a1 = *(const v2f*)(p1d + (k - H_));
        } else {
            a0 = *(const v2f*)(p0e + (k - 2 * H_));
            a1 = *(const v2f*)(p1e + (k - 2 * H_));
        }
        const v2f* bp = Wp1 + (size_t)q * 8 * 32 + lane;
#pragma unroll
        for (int t = 0; t < 8; ++t) {
            const v2f b = bp[t * 32];
            clo[t] = WMMA_F32X4(a0, b, clo[t]);
            chi[t] = WMMA_F32X4(a1, b, chi[t]);
        }
    }
    // bias + relu -> LDS h1 (row pad 132)
#pragma unroll
    for (int t = 0; t < 8; ++t) {
        const int n = t * 16 + m;
        const float bv = bm1[n];
#pragma unroll
        for (int j = 0; j < 8; ++j) {
            const int mm = j + 4 * kh2;
            h1[mm * 132 + n] = fmaxf(clo[t][j] + bv, 0.0f);
            h1[(mm + 16) * 132 + n] = fmaxf(chi[t][j] + bv, 0.0f);
        }
    }
    __syncthreads();

    // ---- layer 2: C2[32x64] = h1[32x128] @ Wm2[128x64] ----
    v8f d2lo[4] = {}, d2hi[4] = {};
    for (int q = 0; q < 32; ++q) {
        const int k = q * 4 + kh2;
        v2f a0 = *(const v2f*)(h1 + m * 132 + k);
        v2f a1 = *(const v2f*)(h1 + (m + 16) * 132 + k);
        const v2f* bp = Wp2 + (size_t)q * 4 * 32 + lane;
#pragma unroll
        for (int t = 0; t < 4; ++t) {
            const v2f b = bp[t * 32];
            d2lo[t] = WMMA_F32X4(a0, b, d2lo[t]);
            d2hi[t] = WMMA_F32X4(a1, b, d2hi[t]);
        }
    }
#pragma unroll
    for (int t = 0; t < 4; ++t) {
        const int n = t * 16 + m;
        const float bv = bm2[n];
#pragma unroll
        for (int j = 0; j < 8; ++j) {
            const int mm = j + 4 * kh2;
            h2[mm * 68 + n] = fmaxf(d2lo[t][j] + bv, 0.0f);
            h2[(mm + 16) * 68 + n] = fmaxf(d2hi[t][j] + bv, 0.0f);
        }
    }
    __syncthreads();

    // ---- layer 3: out[32x2] = h2[32x64] @ Wm3[64x2] + bm3 (one edge/lane) ----
    {
        float acc0 = bm3[0], acc1 = bm3[1];
        const float* hr = h2 + lane * 68;
#pragma unroll 8
        for (int k = 0; k < 64; ++k) {
            const float hv = hr[k];
            acc0 = fmaf(hv, Wm3[2 * k + 0], acc0);
            acc1 = fmaf(hv, Wm3[2 * k + 1], acc1);
        }
        float* op = Out + (size_t)(e0 + lane) * OUT_;
        op[0] = acc0;
        op[1] = acc1;
    }
}

// ---------------------------------------------------------------------------
extern "C" void kernel_launch(void* const* d_in, const int* in_sizes, int n_in,
                              void* d_out, int out_size, void* d_ws, size_t ws_size,
                              hipStream_t stream) {
    const float* x     = (const float*)d_in[0];
    const int*   ei    = (const int*)d_in[1];
    const float* eattr = (const float*)d_in[2];
    const float* Wc    = (const float*)d_in[3];
    const float* bc    = (const float*)d_in[4];
    const float* gamma = (const float*)d_in[5];
    const float* beta  = (const float*)d_in[6];
    const float* rmean = (const float*)d_in[7];
    const float* rvar  = (const float*)d_in[8];
    const float* Wm1   = (const float*)d_in[9];
    const float* bm1   = (const float*)d_in[10];
    const float* Wm2   = (const float*)d_in[11];
    const float* bm2   = (const float*)d_in[12];
    const float* Wm3   = (const float*)d_in[13];
    const float* bm3   = (const float*)d_in[14];
    float* out = (float*)d_out;

    const int* src = ei;
    const int* dst = ei + E_;

    // workspace layout (floats; base is 256B-aligned, all offsets even)
    float* ws   = (float*)d_ws;
    float* deg  = ws;                         // N
    float* dinv = deg + N_;                   // N
    float* xbuf = dinv + N_;                  // N*H
    float* hbuf = xbuf + (size_t)N_ * H_;     // N*H
    float* agg  = hbuf + (size_t)N_ * H_;     // N*H
    v2f* Wcp = (v2f*)(agg + (size_t)N_ * H_); // 3 * 8192 v2f
    v2f* Wp1 = Wcp + 3 * 8192;                // 16896 v2f
    v2f* Wp2 = Wp1 + 16896;                   // 4096 v2f

    // degree + normalization
    hipMemsetAsync(deg, 0, (size_t)N_ * sizeof(float), stream);
    deg_kernel<<<(E_ + 255) / 256, 256, 0, stream>>>(dst, deg);
    dinv_kernel<<<(N_ + 255) / 256, 256, 0, stream>>>(deg, dinv);

    // pack all weights into B-fragment order
    for (int l = 0; l < 3; ++l)
        pack_b_kernel<<<(8192 + 255) / 256, 256, 0, stream>>>(
            Wc + (size_t)l * D_ * H_, Wcp + (size_t)l * 8192, 128, 128);
    pack_b_kernel<<<(16896 + 255) / 256, 256, 0, stream>>>(Wm1, Wp1, 264, 128);
    pack_b_kernel<<<(4096 + 255) / 256, 256, 0, stream>>>(Wm2, Wp2, 128, 64);

    // x -> xbuf (do not mutate inputs)
    hipMemcpyAsync(xbuf, x, (size_t)N_ * D_ * sizeof(float),
                   hipMemcpyDeviceToDevice, stream);

    const int gemm_blocks = (N_ / 16 + 7) / 8;        // 391
    for (int l = 0; l < 3; ++l) {
        gemm_node_kernel<<<gemm_blocks, 256, 0, stream>>>(
            xbuf, Wcp + (size_t)l * 8192, hbuf);
        hipMemsetAsync(agg, 0, (size_t)N_ * H_ * sizeof(float), stream);
        scatter_kernel<<<(E_ * 32) / 256, 256, 0, stream>>>(hbuf, src, dst, dinv, agg);
        combine_kernel<<<(N_ * H_ + 255) / 256, 256, 0, stream>>>(
            agg, hbuf, dinv, bc + l * H_, gamma + l * H_, beta + l * H_,
            rmean + l * H_, rvar + l * H_, xbuf);
    }

    // fused edge-level MLP head: 6250 blocks x 4 waves x 32 edges
    const int eh_blocks = E_ / (4 * 32);
    const size_t eh_lds = 4 * 6400 * sizeof(float);   // 100 KB / workgroup
    edge_head_kernel<<<eh_blocks, 128, eh_lds, stream>>>(
        xbuf, src, dst, eattr, Wp1, bm1, Wp2, bm2, Wm3, bm3, out);
}